// TwoGATCN_87720412053583
// MI455X (gfx1250) — compile-verified
//
#include <hip/hip_runtime.h>
#include <hip/hip_bf16.h>
#include <math.h>

typedef float v2f __attribute__((ext_vector_type(2)));
typedef float v4f __attribute__((ext_vector_type(4)));
typedef float v8f __attribute__((ext_vector_type(8)));

#define NEG_SLOPE 0.2f
#define SCAN_BS 256

// ---------------------------------------------------------------------------
// Fast FP32 WMMA GEMM:  C[M x N] = A[M x K] @ B[K x N],  N % (TN*16) == 0.
// One wave32 computes a 16 x (TN*16) block via V_WMMA_F32_16X16X4_F32.
//   A frag (16x4): lane l -> row m=l%16; holds K = kg,kg+1, kg = 2*(l>=16).
//   B frag (4x16): lane l -> col n=l%16; holds K = kg,kg+1.
//   C (16x16): lane l -> col n=l%16; VGPR j -> row j + 8*(l>=16).
// M must be a multiple of 16 (n = 50000 = 3125*16).
// ---------------------------------------------------------------------------
template <int TN>
__global__ void gemm_wmma_f32_fast(const float* __restrict__ A,
                                   const float* __restrict__ B,
                                   float* __restrict__ C,
                                   int K, int N, int Mt, int Ng)
{
    int wave = blockIdx.x * (blockDim.x >> 5) + (threadIdx.x >> 5);
    if (wave >= Mt * Ng) return;          // whole-wave exit, EXEC stays all-1s
    int mt = wave / Ng;
    int ng = wave % Ng;

    int lane = threadIdx.x & 31;
    int lm   = lane & 15;
    int kg   = (lane >> 4) << 1;          // 0 or 2

    const float* ap  = A + (size_t)(mt * 16 + lm) * K + kg;
    const float* bp0 = B + (size_t)kg * N + ng * (TN * 16) + lm;  // row kg
    const float* bp1 = bp0 + N;                                   // row kg+1
    const size_t bstep = (size_t)4 * N;

    v8f acc[TN];
    #pragma unroll
    for (int t = 0; t < TN; ++t) acc[t] = (v8f){};

    #pragma unroll 2
    for (int k0 = 0; k0 < K; k0 += 4) {
        v2f a;
        a.x = ap[0];
        a.y = ap[1];
        ap += 4;
        #pragma unroll
        for (int t = 0; t < TN; ++t) {
            v2f b;
            b.x = bp0[t * 16];
            b.y = bp1[t * 16];
            acc[t] = __builtin_amdgcn_wmma_f32_16x16x4_f32(
                         false, a, false, b, (short)0, acc[t], false, false);
        }
        bp0 += bstep;
        bp1 += bstep;
    }

    int rowBase = mt * 16 + ((lane >> 4) << 3);   // +0 or +8
    #pragma unroll
    for (int t = 0; t < TN; ++t) {
        float* cp = C + (size_t)rowBase * N + ng * (TN * 16) + t * 16 + lm;
        #pragma unroll
        for (int j = 0; j < 8; ++j)
            cp[(size_t)j * N] = acc[t][j];
    }
}

// ---------------------------------------------------------------------------
// Ragged-N FP32 WMMA GEMM (+bias) for the classifier (Nact = 40).
// WMMA columns are independent, so out-of-range B columns are only
// address-clamped; their outputs are never stored. No divergence in K-loop.
// ---------------------------------------------------------------------------
__global__ void gemm_wmma_f32_ragged(const float* __restrict__ A,
                                     const float* __restrict__ B,
                                     float* __restrict__ C,
                                     int K, int Nact, int Mt, int Nt,
                                     const float* __restrict__ bias)
{
    int wave = blockIdx.x * (blockDim.x >> 5) + (threadIdx.x >> 5);
    if (wave >= Mt * Nt) return;
    int mt = wave / Nt;
    int nt = wave % Nt;

    int lane = threadIdx.x & 31;
    int lm   = lane & 15;
    int kg   = (lane >> 4) << 1;

    int col  = nt * 16 + lm;
    int colc = (col < Nact) ? col : (Nact - 1);     // clamped, in-bounds

    const float* ap  = A + (size_t)(mt * 16 + lm) * K + kg;
    const float* bp0 = B + (size_t)kg * Nact + colc;
    const float* bp1 = bp0 + Nact;
    const size_t bstep = (size_t)4 * Nact;

    v8f acc = {};
    #pragma unroll 2
    for (int k0 = 0; k0 < K; k0 += 4) {
        v2f a, b;
        a.x = ap[0];
        a.y = ap[1];
        ap += 4;
        b.x = bp0[0];
        b.y = bp1[0];
        bp0 += bstep;
        bp1 += bstep;
        acc = __builtin_amdgcn_wmma_f32_16x16x4_f32(
                  false, a, false, b, (short)0, acc, false, false);
    }

    if (col < Nact) {
        float bv = bias ? bias[col] : 0.f;
        int rowBase = mt * 16 + ((lane >> 4) << 3);
        #pragma unroll
        for (int j = 0; j < 8; ++j)
            C[(size_t)(rowBase + j) * Nact + col] = acc[j] + bv;
    }
}

// ---------------------------------------------------------------------------
// CSR build (once per launch; graph shared by both layers).
// Edge id space: e in [0, E) are ei edges, e in [E, E+n) are self loops.
// ---------------------------------------------------------------------------
__global__ void zero_int(int* __restrict__ p, int cnt)
{
    int i = blockIdx.x * blockDim.x + threadIdx.x;
    if (i < cnt) p[i] = 0;
}

__global__ void count_deg(const int* __restrict__ ei, int E, int n,
                          int* __restrict__ cnt)
{
    int e = blockIdx.x * blockDim.x + threadIdx.x;
    if (e >= E + n) return;
    int d = (e < E) ? ei[E + e] : (e - E);
    atomicAdd(&cnt[d], 1);
}

// exclusive scan, stage 1: per-block scan + block sums
__global__ void scan_block(const int* __restrict__ in, int* __restrict__ out,
                           int* __restrict__ bsum, int n)
{
    __shared__ int sh[SCAN_BS];
    int gid = blockIdx.x * SCAN_BS + threadIdx.x;
    int v = (gid < n) ? in[gid] : 0;
    sh[threadIdx.x] = v;
    __syncthreads();
    for (int off = 1; off < SCAN_BS; off <<= 1) {
        int t = (threadIdx.x >= off) ? sh[threadIdx.x - off] : 0;
        __syncthreads();
        sh[threadIdx.x] += t;
        __syncthreads();
    }
    if (gid < n) out[gid] = sh[threadIdx.x] - v;          // exclusive
    if (threadIdx.x == SCAN_BS - 1) bsum[blockIdx.x] = sh[threadIdx.x];
}

// stage 2: single block scans the block sums (handles any nb via carry loop)
__global__ void scan_bsums(int* __restrict__ bsum, int nb)
{
    __shared__ int sh[SCAN_BS];
    int carry = 0;
    for (int base = 0; base < nb; base += SCAN_BS) {
        int idx = base + threadIdx.x;
        int v = (idx < nb) ? bsum[idx] : 0;
        sh[threadIdx.x] = v;
        __syncthreads();
        for (int off = 1; off < SCAN_BS; off <<= 1) {
            int t = (threadIdx.x >= off) ? sh[threadIdx.x - off] : 0;
            __syncthreads();
            sh[threadIdx.x] += t;
            __syncthreads();
        }
        int total = sh[SCAN_BS - 1];
        if (idx < nb) bsum[idx] = sh[threadIdx.x] - v + carry;
        carry += total;
        __syncthreads();
    }
}

// stage 3: add block offsets; rowptr[n] = total edge count
__global__ void scan_add(int* __restrict__ rowptr, const int* __restrict__ bsum,
                         int n, int Et)
{
    int gid = blockIdx.x * SCAN_BS + threadIdx.x;
    if (gid < n) rowptr[gid] += bsum[blockIdx.x];
    if (gid == 0) rowptr[n] = Et;
}

__global__ void copy_int(const int* __restrict__ src, int* __restrict__ dst, int cnt)
{
    int i = blockIdx.x * blockDim.x + threadIdx.x;
    if (i < cnt) dst[i] = src[i];
}

__global__ void fill_csr(const int* __restrict__ ei, int E, int n,
                         int* __restrict__ cursor, int* __restrict__ eidx)
{
    int e = blockIdx.x * blockDim.x + threadIdx.x;
    if (e >= E + n) return;
    int d = (e < E) ? ei[E + e] : (e - E);
    int pos = atomicAdd(&cursor[d], 1);
    eidx[pos] = e;
}

// ---------------------------------------------------------------------------
// Pass A: logit[e] = att . leaky_relu(xl[src] + xr[dst]).
// One wave per edge; float4 gathers (Cdim % 4 == 0).
// ---------------------------------------------------------------------------
__global__ void edge_logits(const float* __restrict__ xl,
                            const float* __restrict__ xr,
                            const float* __restrict__ att,
                            const int* __restrict__ ei,
                            int E, int n, int Cdim,
                            float* __restrict__ logits)
{
    int e = blockIdx.x * (blockDim.x >> 5) + (threadIdx.x >> 5);
    if (e >= E + n) return;
    int lane = threadIdx.x & 31;
    int s, d;
    if (e < E) { s = ei[e]; d = ei[E + e]; }
    else       { s = e - E; d = s; }          // self loops appended last

    const v4f* ps = (const v4f*)(xl + (size_t)s * Cdim);
    const v4f* pd = (const v4f*)(xr + (size_t)d * Cdim);
    const v4f* pa = (const v4f*)att;
    int nv = Cdim >> 2;

    float t = 0.f;
    for (int c = lane; c < nv; c += 32) {
        v4f vs = ps[c];
        v4f vd = pd[c];
        v4f va = pa[c];
        #pragma unroll
        for (int q = 0; q < 4; ++q) {
            float v = vs[q] + vd[q];
            v = (v > 0.f) ? v : NEG_SLOPE * v;
            t = fmaf(va[q], v, t);
        }
    }
    #pragma unroll
    for (int off = 16; off > 0; off >>= 1)
        t += __shfl_xor(t, off, 32);

    if (lane == 0) logits[e] = t;
}

// ---------------------------------------------------------------------------
// Per-node softmax + gather-aggregate, fused with bias+ReLU. No atomics.
// One wave per node; R = Cdim/128 float4 accumulators per lane.
//   h[i] = relu( (sum_j exp(lg_j - max) * xl[src_j]) / sum_j exp(lg_j - max) + b )
// ---------------------------------------------------------------------------
template <int R>
__global__ void node_aggregate(const float* __restrict__ xl,
                               const float* __restrict__ logits,
                               const int* __restrict__ rowptr,
                               const int* __restrict__ eidx,
                               const int* __restrict__ ei,
                               int E, int n, int Cdim,
                               const float* __restrict__ bias,
                               float* __restrict__ h)
{
    int i = blockIdx.x * (blockDim.x >> 5) + (threadIdx.x >> 5);
    if (i >= n) return;
    int lane = threadIdx.x & 31;
    int beg = rowptr[i];
    int end = rowptr[i + 1];

    // segment max, lane-parallel (self loop guarantees end > beg)
    float mx = -INFINITY;
    for (int j = beg + lane; j < end; j += 32)
        mx = fmaxf(mx, logits[eidx[j]]);
    #pragma unroll
    for (int off = 16; off > 0; off >>= 1)
        mx = fmaxf(mx, __shfl_xor(mx, off, 32));

    v4f acc[R];
    #pragma unroll
    for (int r = 0; r < R; ++r) acc[r] = (v4f){};
    float dsum = 0.f;

    for (int j = beg; j < end; ++j) {
        int eid = eidx[j];
        float a = __expf(logits[eid] - mx);   // uniform across lanes
        dsum += a;
        int s = (eid < E) ? ei[eid] : (eid - E);
        const v4f* ps = (const v4f*)(xl + (size_t)s * Cdim);
        #pragma unroll
        for (int r = 0; r < R; ++r) {
            v4f v = ps[lane + r * 32];
            #pragma unroll
            for (int q = 0; q < 4; ++q)
                acc[r][q] = fmaf(a, v[q], acc[r][q]);
        }
    }

    float inv = 1.f / dsum;
    #pragma unroll
    for (int r = 0; r < R; ++r) {
        int c = (lane + r * 32) * 4;
        v4f bv = *(const v4f*)(bias + c);
        v4f o;
        #pragma unroll
        for (int q = 0; q < 4; ++q) {
            float v = fmaf(acc[r][q], inv, bv[q]);
            o[q] = v > 0.f ? v : 0.f;
        }
        *(v4f*)(h + (size_t)i * Cdim + c) = o;
    }
}

// ---------------------------------------------------------------------------
static inline void launch_gemm(const float* A, const float* B, float* C,
                               int M, int K, int N, const float* bias,
                               hipStream_t stream)
{
    int Mt = M / 16;
    if ((N & 63) == 0 && bias == nullptr) {
        int Ng = N / 64;
        int blocks = (Mt * Ng + 3) / 4;
        gemm_wmma_f32_fast<4><<<blocks, 128, 0, stream>>>(A, B, C, K, N, Mt, Ng);
    } else {
        int Nt = (N + 15) / 16;
        int blocks = (Mt * Nt + 3) / 4;
        gemm_wmma_f32_ragged<<<blocks, 128, 0, stream>>>(A, B, C, K, N, Mt, Nt, bias);
    }
}

static inline void launch_aggregate(const float* xl, const float* lg,
                                    const int* rowptr, const int* eidx,
                                    const int* ei, int E, int n, int Cdim,
                                    const float* bias, float* h,
                                    hipStream_t stream)
{
    int blocks = (n + 7) / 8;   // 8 waves / 256-thread block
    switch (Cdim / 128) {
        case 1: node_aggregate<1><<<blocks, 256, 0, stream>>>(xl, lg, rowptr, eidx, ei, E, n, Cdim, bias, h); break;
        case 2: node_aggregate<2><<<blocks, 256, 0, stream>>>(xl, lg, rowptr, eidx, ei, E, n, Cdim, bias, h); break;
        case 3: node_aggregate<3><<<blocks, 256, 0, stream>>>(xl, lg, rowptr, eidx, ei, E, n, Cdim, bias, h); break;
        default: node_aggregate<4><<<blocks, 256, 0, stream>>>(xl, lg, rowptr, eidx, ei, E, n, Cdim, bias, h); break;
    }
}

extern "C" void kernel_launch(void* const* d_in, const int* in_sizes, int n_in,
                              void* d_out, int out_size, void* d_ws, size_t ws_size,
                              hipStream_t stream)
{
    const float* x    = (const float*)d_in[0];
    const int*   ei   = (const int*)  d_in[1];
    const float* W1l  = (const float*)d_in[2];
    const float* W1r  = (const float*)d_in[3];
    const float* att1 = (const float*)d_in[4];
    const float* b1   = (const float*)d_in[5];
    const float* W2l  = (const float*)d_in[6];
    const float* W2r  = (const float*)d_in[7];
    const float* att2 = (const float*)d_in[8];
    const float* b2   = (const float*)d_in[9];
    const float* Wc   = (const float*)d_in[10];
    const float* bc   = (const float*)d_in[11];
    float* out = (float*)d_out;

    const int H1 = in_sizes[4];            // 256
    const int H2 = in_sizes[8];            // 128
    const int Cc = in_sizes[11];           // 40
    const int F  = in_sizes[2] / H1;       // 128
    const int n  = in_sizes[0] / F;        // 50000
    const int E  = in_sizes[1] / 2;        // 800000
    const int Et = E + n;
    const int nb = (n + SCAN_BS - 1) / SCAN_BS;

    // ---- workspace layout (4-byte words); h aliases xr (aggregate reads xl)
    float* f   = (float*)d_ws;
    size_t off = 0;
    float* xl1 = f + off; off += (size_t)n * H1;
    float* xr1 = f + off; off += (size_t)n * H1;     // later: h1
    float* xl2 = f + off; off += (size_t)n * H2;
    float* xr2 = f + off; off += (size_t)n * H2;     // later: h2
    float* lg  = f + off; off += (size_t)Et;
    int* rowptr = (int*)(f + off); off += (size_t)n + 1;
    int* eidx   = (int*)(f + off); off += (size_t)Et;
    int* work   = (int*)(f + off); off += (size_t)n;   // degree counts / cursor
    int* bsum   = (int*)(f + off); off += (size_t)nb;

    const int edgeBlocks  = (Et + 7) / 8;              // 8 waves/block
    const int edgeThreads = (Et + 255) / 256;          // thread-per-edge grids

    // ---------------- CSR build (once; shared by both layers) ----------------
    zero_int<<<(n + 255) / 256, 256, 0, stream>>>(work, n);
    count_deg<<<edgeThreads, 256, 0, stream>>>(ei, E, n, work);
    scan_block<<<nb, SCAN_BS, 0, stream>>>(work, rowptr, bsum, n);
    scan_bsums<<<1, SCAN_BS, 0, stream>>>(bsum, nb);
    scan_add<<<nb, SCAN_BS, 0, stream>>>(rowptr, bsum, n, Et);
    copy_int<<<(n + 255) / 256, 256, 0, stream>>>(rowptr, work, n);
    fill_csr<<<edgeThreads, 256, 0, stream>>>(ei, E, n, work, eidx);

    // ---------------- layer 1 ----------------
    launch_gemm(x, W1l, xl1, n, F, H1, nullptr, stream);
    launch_gemm(x, W1r, xr1, n, F, H1, nullptr, stream);
    edge_logits<<<edgeBlocks, 256, 0, stream>>>(xl1, xr1, att1, ei, E, n, H1, lg);
    launch_aggregate(xl1, lg, rowptr, eidx, ei, E, n, H1, b1, xr1, stream);   // h1 = xr1

    // ---------------- layer 2 ----------------
    launch_gemm(xr1, W2l, xl2, n, H1, H2, nullptr, stream);
    launch_gemm(xr1, W2r, xr2, n, H1, H2, nullptr, stream);
    edge_logits<<<edgeBlocks, 256, 0, stream>>>(xl2, xr2, att2, ei, E, n, H2, lg);
    launch_aggregate(xl2, lg, rowptr, eidx, ei, E, n, H2, b2, xr2, stream);   // h2 = xr2

    // ---------------- classifier ----------------
    launch_gemm(xr2, Wc, out, n, H2, Cc, bc, stream);
}